// WindowAttention_6734508720191
// MI455X (gfx1250) — compile-verified
//
#include <hip/hip_runtime.h>

typedef float v2f __attribute__((ext_vector_type(2)));
typedef float v8f __attribute__((ext_vector_type(8)));

#define NHEAD   16
#define HDIM    32
#define CDIM    512
#define LWIN    49          // 7*7 tokens per window
#define SCALE   0.17677669529663687f   // 1/sqrt(32)

// D = A(16x4) * B(4x16) + C(16x16), fp32, wave32
__device__ __forceinline__ v8f wmma4(v2f a, v2f b, v8f c) {
    return __builtin_amdgcn_wmma_f32_16x16x4_f32(
        /*neg_a=*/false, a, /*neg_b=*/false, b,
        /*c_mod=*/(short)0, c, /*reuse_a=*/false, /*reuse_b=*/false);
}

// ---------------------------------------------------------------------------
// Kernel 1: fused QKV projection + window attention for one (window, head).
// grid.x = 2048 * 16 (win-major so the 16 head-blocks of a window share L2/x),
// block = 128 threads (4 waves; wave w owns output rows 16w..16w+15 of M=64).
// WMMA fragments are read from cooperatively staged LDS; all but GEMM3's B by ds_load_b64.
// ---------------------------------------------------------------------------
__launch_bounds__(128)
__global__ void qkv_attn_kernel(const float* __restrict__ x,
                                const float* __restrict__ mask,
                                const float* __restrict__ w_qkv,
                                const float* __restrict__ b_qkv,
                                const float* __restrict__ bias_table,
                                const int*   __restrict__ rel_index,
                                float* __restrict__ attn_out) {
    // qs/ks/vs: 64x36 each (stride 36 -> conflict-free 16-row fragment reads)
    __shared__ float qs[64 * 36];
    __shared__ float ks_[64 * 36];
    __shared__ float vs[64 * 36];
    // staging union: xs[64][36] + wsl_t[96][34] (k-chunk)  OR  ss[64][68]
    __shared__ __attribute__((aligned(16))) float ubuf[64 * 36 + 96 * 34]; // 22272 B
    float* xs    = ubuf;              // x chunk  [64 rows][32 k + pad], stride 36
    float* wsl_t = ubuf + 64 * 36;    // w chunk transposed [96 cols][32 k + pad], stride 34

    const int tid  = threadIdx.x;
    const int lane = tid & 31;
    const int wid  = tid >> 5;
    const int grp  = lane >> 4;       // lane group: selects K pair / upper M rows
    const int ln   = lane & 15;

    const int blk  = blockIdx.x;
    const int head = blk & (NHEAD - 1);
    const int win  = blk >> 4;        // 0..2047
    const int bimg = win >> 4;
    const int widx = win & 15;        // window index within image (mask index)
    const int hW   = widx >> 2;
    const int wW   = widx & 3;

    const int m0  = wid * 16;
    const int cb0 = head * HDIM;

    // ---- GEMM1: qkv = x_window @ w_qkv (96 cols for this head), K=512 ------
    v8f acc[6];
    #pragma unroll
    for (int j = 0; j < 6; ++j)
        #pragma unroll
        for (int e = 0; e < 8; ++e) acc[j][e] = 0.0f;

    for (int kc = 0; kc < 16; ++kc) {          // K chunks of 32
        __syncthreads();
        // stage x chunk [64 x 32] -> xs (rows >= 49 zeroed)
        for (int idx = tid; idx < 64 * 8; idx += 128) {
            int row = idx >> 3;
            int c4  = (idx & 7) << 2;
            float4 val = make_float4(0.f, 0.f, 0.f, 0.f);
            if (row < LWIN) {
                int h = hW * 7 + row / 7;
                int w = wW * 7 + row % 7;
                const float* src = x + (((size_t)bimg * 28 + h) * 28 + w) * CDIM
                                     + kc * 32 + c4;
                val = *(const float4*)src;
            }
            *(float4*)&xs[row * 36 + c4] = val;
        }
        // stage w_qkv chunk [32 k x 96 cols] transposed -> wsl_t[col][k]
        for (int idx = tid; idx < 32 * 24; idx += 128) {
            int krow = idx / 24;
            int t    = idx % 24;
            int seg  = t >> 3;                 // 0=q 1=k 2=v
            int c4   = (t & 7) << 2;
            const float* src = w_qkv + (size_t)(kc * 32 + krow) * (3 * CDIM)
                                     + seg * CDIM + cb0 + c4;
            float4 v4 = *(const float4*)src;
            wsl_t[(seg * 32 + c4 + 0) * 34 + krow] = v4.x;
            wsl_t[(seg * 32 + c4 + 1) * 34 + krow] = v4.y;
            wsl_t[(seg * 32 + c4 + 2) * 34 + krow] = v4.z;
            wsl_t[(seg * 32 + c4 + 3) * 34 + krow] = v4.w;
        }
        __syncthreads();

        for (int k = 0; k < 32; k += 4) {
            v2f a = *(const v2f*)&xs[(m0 + ln) * 36 + k + 2 * grp];
            #pragma unroll
            for (int j = 0; j < 6; ++j) {
                int col = (j >> 1) * 32 + (j & 1) * 16 + ln;
                v2f b = *(const v2f*)&wsl_t[col * 34 + k + 2 * grp];
                acc[j] = wmma4(a, b, acc[j]);
            }
        }
    }

    // store q (scaled), k, v fragments to LDS (+bias)
    #pragma unroll
    for (int j = 0; j < 6; ++j) {
        int which = j >> 1;                     // 0=q 1=k 2=v
        float* dst = which == 0 ? qs : (which == 1 ? ks_ : vs);
        int col = (j & 1) * 16 + ln;
        float bias = b_qkv[cb0 + which * CDIM + col];
        #pragma unroll
        for (int r = 0; r < 8; ++r) {
            int row = m0 + r + 8 * grp;
            float v = acc[j][r] + bias;
            if (which == 0) v *= SCALE;
            dst[row * 36 + col] = v;
        }
    }
    __syncthreads();

    // ---- GEMM2: S = q @ k^T  (M=64, N=64, K=32) ----------------------------
    float* ss = ubuf; // stride 68
    {
        v8f s_acc[4];
        #pragma unroll
        for (int nt = 0; nt < 4; ++nt)
            #pragma unroll
            for (int e = 0; e < 8; ++e) s_acc[nt][e] = 0.0f;

        for (int k = 0; k < 32; k += 4) {
            v2f a = *(const v2f*)&qs[(m0 + ln) * 36 + k + 2 * grp];
            #pragma unroll
            for (int nt = 0; nt < 4; ++nt) {
                // B[kdim][n] = k[n][kdim] -> contiguous pair in ks_ row n
                v2f b = *(const v2f*)&ks_[(nt * 16 + ln) * 36 + k + 2 * grp];
                s_acc[nt] = wmma4(a, b, s_acc[nt]);
            }
        }
        // + relative position bias + shifted-window mask, write S to LDS
        #pragma unroll
        for (int nt = 0; nt < 4; ++nt) {
            #pragma unroll
            for (int r = 0; r < 8; ++r) {
                int l1 = m0 + r + 8 * grp;
                int l2 = nt * 16 + ln;
                float v = 0.0f;
                if (l1 < LWIN && l2 < LWIN) {
                    int ri = rel_index[l1 * LWIN + l2];
                    v = s_acc[nt][r] + bias_table[ri * NHEAD + head]
                        + mask[((size_t)widx * LWIN + l1) * LWIN + l2];
                }
                ss[l1 * 68 + l2] = v;   // padded rows/cols zeroed
            }
        }
    }
    __syncthreads();

    // ---- softmax over the 49 valid columns of each valid row ---------------
    if (tid < LWIN) {
        float* row = ss + tid * 68;
        float mx = -1e30f;
        for (int j = 0; j < LWIN; ++j) mx = fmaxf(mx, row[j]);
        float sum = 0.0f;
        for (int j = 0; j < LWIN; ++j) {
            float e = expf(row[j] - mx);
            row[j] = e;
            sum += e;
        }
        float inv = 1.0f / sum;
        for (int j = 0; j < LWIN; ++j) row[j] *= inv;
        // cols 49..63 remain exactly 0 -> K=64 padding in GEMM3 is exact
    }
    __syncthreads();

    // ---- GEMM3: out = P @ v  (M=64, N=32, K=64) ----------------------------
    v8f o_acc[2];
    #pragma unroll
    for (int nt = 0; nt < 2; ++nt)
        #pragma unroll
        for (int e = 0; e < 8; ++e) o_acc[nt][e] = 0.0f;

    for (int k = 0; k < 64; k += 4) {
        v2f a = *(const v2f*)&ss[(m0 + ln) * 68 + k + 2 * grp];
        #pragma unroll
        for (int nt = 0; nt < 2; ++nt) {
            v2f b;
            b.x = vs[(k + 2 * grp) * 36 + nt * 16 + ln];
            b.y = vs[(k + 2 * grp + 1) * 36 + nt * 16 + ln];
            o_acc[nt] = wmma4(a, b, o_acc[nt]);
        }
    }
    // scatter to window-reversed [B,28,28,C] layout in workspace
    #pragma unroll
    for (int nt = 0; nt < 2; ++nt) {
        #pragma unroll
        for (int r = 0; r < 8; ++r) {
            int l = m0 + r + 8 * grp;
            if (l < LWIN) {
                int h = hW * 7 + l / 7;
                int w = wW * 7 + l % 7;
                int c = head * HDIM + nt * 16 + ln;
                attn_out[(((size_t)bimg * 28 + h) * 28 + w) * CDIM + c] = o_acc[nt][r];
            }
        }
    }
}

// ---------------------------------------------------------------------------
// Kernel 2: output projection GEMM  [100352,512] x [512,512] + bias.
// 128-thread blocks computing a 64x128 C macro-tile; A and B k-chunks staged
// in LDS (B transposed), every fragment a ds_load_b64; A reused over 8 N-tiles.
// ---------------------------------------------------------------------------
__launch_bounds__(128)
__global__ void proj_kernel(const float* __restrict__ a_in,
                            const float* __restrict__ w_proj,
                            const float* __restrict__ b_proj,
                            float* __restrict__ out) {
    __shared__ __attribute__((aligned(16))) float xs[64 * 36];   // A chunk [64][32+pad]
    __shared__ float wt[128 * 34];                               // B^T chunk [128 cols][32+pad]

    const int tid  = threadIdx.x;
    const int lane = tid & 31;
    const int wid  = tid >> 5;
    const int grp  = lane >> 4;
    const int ln   = lane & 15;

    const size_t m0 = (size_t)blockIdx.x * 64;
    const int    n0 = blockIdx.y * 128;

    v8f acc[8];
    #pragma unroll
    for (int nt = 0; nt < 8; ++nt)
        #pragma unroll
        for (int e = 0; e < 8; ++e) acc[nt][e] = 0.0f;

    for (int kc = 0; kc < 16; ++kc) {          // K chunks of 32
        __syncthreads();
        // stage A chunk [64 x 32]
        for (int idx = tid; idx < 64 * 8; idx += 128) {
            int row = idx >> 3;
            int c4  = (idx & 7) << 2;
            float4 v4 = *(const float4*)&a_in[(m0 + row) * CDIM + kc * 32 + c4];
            *(float4*)&xs[row * 36 + c4] = v4;
        }
        // stage B chunk [32 k x 128 cols] transposed -> wt[col][k]
        for (int idx = tid; idx < 32 * 32; idx += 128) {
            int krow = idx >> 5;
            int c4   = (idx & 31) << 2;
            float4 v4 = *(const float4*)&w_proj[(size_t)(kc * 32 + krow) * CDIM + n0 + c4];
            wt[(c4 + 0) * 34 + krow] = v4.x;
            wt[(c4 + 1) * 34 + krow] = v4.y;
            wt[(c4 + 2) * 34 + krow] = v4.z;
            wt[(c4 + 3) * 34 + krow] = v4.w;
        }
        __syncthreads();

        for (int k = 0; k < 32; k += 4) {
            v2f a = *(const v2f*)&xs[(wid * 16 + ln) * 36 + k + 2 * grp];
            #pragma unroll
            for (int nt = 0; nt < 8; ++nt) {
                v2f b = *(const v2f*)&wt[(nt * 16 + ln) * 34 + k + 2 * grp];
                acc[nt] = wmma4(a, b, acc[nt]);
            }
        }
    }

    #pragma unroll
    for (int nt = 0; nt < 8; ++nt) {
        float bias = b_proj[n0 + nt * 16 + ln];
        #pragma unroll
        for (int r = 0; r < 8; ++r) {
            size_t row = m0 + wid * 16 + r + 8 * grp;
            out[row * CDIM + n0 + nt * 16 + ln] = acc[nt][r] + bias;
        }
    }
}

// ---------------------------------------------------------------------------
extern "C" void kernel_launch(void* const* d_in, const int* in_sizes, int n_in,
                              void* d_out, int out_size, void* d_ws, size_t ws_size,
                              hipStream_t stream) {
    const float* x          = (const float*)d_in[0];
    const float* mask       = (const float*)d_in[1];
    const float* w_qkv      = (const float*)d_in[2];
    const float* b_qkv      = (const float*)d_in[3];
    const float* bias_table = (const float*)d_in[4];
    const float* w_proj     = (const float*)d_in[5];
    const float* b_proj     = (const float*)d_in[6];
    const int*   rel_index  = (const int*)d_in[7];
    // d_in[8]=height, d_in[9]=width are compile-time constants here (28, 28)

    float* out     = (float*)d_out;
    float* attn_ws = (float*)d_ws;     // [128,28,28,512] fp32 = 205.5 MB

    // 2048 windows * 16 heads, win-major so head-blocks of a window co-schedule
    qkv_attn_kernel<<<2048 * 16, 128, 0, stream>>>(
        x, mask, w_qkv, b_qkv, bias_table, rel_index, attn_ws);

    // 100352/64 = 1568 row tiles, 512/128 = 4 col tiles
    proj_kernel<<<dim3(1568, 4), 128, 0, stream>>>(attn_ws, w_proj, b_proj, out);
}